// CoPE_81922206204605
// MI455X (gfx1250) — compile-verified
//
#include <hip/hip_runtime.h>

// CoPE on MI455X (gfx1250, wave32).
//
// Shapes: q (16,2048,64) f32, qk (16,2048,2048) f32, pos_emb (1,64,64) f32.
// Out:    (16,2048,2048) f32.
//
// Memory-bound: 256MB qk read + 256MB out write => ~22us floor @ 23.3 TB/s.
// Kernel 1: E = q @ pos_emb via V_WMMA_F32_16X16X4_F32 (exact f32 math).
// Kernel 2: per-row reverse cumsum (wave scan) + LDS gather interpolation,
//           streaming with non-temporal loads/stores.

typedef __attribute__((ext_vector_type(2))) float v2f;
typedef __attribute__((ext_vector_type(8))) float v8f;

#define BH 16
#define NN 2048
#define CC 64
#define TT 64

// ---------------------------------------------------------------------------
// Kernel 1: E[row, t] = sum_c q[row, c] * pe[c, t], rows = BH*NN.
// One wave per 16-row tile; D(16x64) = A(16x64) x B(64x64) as 4 n-tiles x 16
// K-steps of V_WMMA_F32_16X16X4_F32.
//
// ISA 32-bit layouts (CDNA5 ISA):
//   A 16x4:  lane<16: VGPR0=A[m][K0], VGPR1=A[m][K0+1];
//            lane>=16: VGPR0=A[m][K0+2], VGPR1=A[m][K0+3]   (m = lane&15)
//   B 4x16:  lane half selects K row pair (K0+0/1 vs K0+2/3), n = lane&15
//   C/D 16x16: VGPR v: lanes 0-15 -> M=v, lanes 16-31 -> M=v+8, N=lane&15
// ---------------------------------------------------------------------------
__global__ void __launch_bounds__(256)
cope_e_wmma(const float* __restrict__ q, const float* __restrict__ pe,
            float* __restrict__ E) {
  const int lane = threadIdx.x & 31;
  const int wave = threadIdx.x >> 5;
  const int tile = blockIdx.x * 8 + wave;  // 16-row tile of E
  const int rowBase = tile * 16;
  const int m = lane & 15;
  const int koff = (lane >> 4) * 2;        // 0 for lanes 0-15, 2 for 16-31

  v8f acc[4];
#pragma unroll
  for (int nt = 0; nt < 4; ++nt) acc[nt] = (v8f){0.f,0.f,0.f,0.f,0.f,0.f,0.f,0.f};

#pragma unroll
  for (int kk = 0; kk < 16; ++kk) {
    const int K0 = kk * 4;
    const float* ap = q + (size_t)(rowBase + m) * CC + (K0 + koff);
    v2f a;
    a.x = ap[0];
    a.y = ap[1];
#pragma unroll
    for (int nt = 0; nt < 4; ++nt) {
      const int t = nt * 16 + m;
      v2f b;
      b.x = pe[(K0 + koff + 0) * TT + t];
      b.y = pe[(K0 + koff + 1) * TT + t];
      // 8 args: (neg_a, A, neg_b, B, c_mod, C, reuse_a, reuse_b)
      acc[nt] = __builtin_amdgcn_wmma_f32_16x16x4_f32(
          false, a, false, b, (short)0, acc[nt], false, false);
    }
  }

  const int rOff = (lane >> 4) * 8;
#pragma unroll
  for (int nt = 0; nt < 4; ++nt) {
    const int col = nt * 16 + (lane & 15);
#pragma unroll
    for (int v = 0; v < 8; ++v) {
      const int r = rowBase + rOff + v;
      E[(size_t)r * TT + col] = acc[nt][v];
    }
  }
}

// ---------------------------------------------------------------------------
// Kernel 2: one wave32 per row of qk.
//   P[j] = sum_{k>=j} sigmoid(qk[row,k])   (suffix sum, computed right-to-left
//   in 32-wide chunks: reversed lane mapping turns an inclusive lane scan into
//   a suffix scan; scalar carry chains the 64 chunks)
//   out[j] = lerp(E[row, floor(Pc)], E[row, ceil(Pc)], frac(Pc)), Pc=min(P,63)
// ---------------------------------------------------------------------------
__global__ void __launch_bounds__(256)
cope_scan(const float* __restrict__ qk, const float* __restrict__ E,
          float* __restrict__ out) {
  __shared__ float eRow[8][TT];
  const int lane = threadIdx.x & 31;
  const int wave = threadIdx.x >> 5;
  const size_t row = (size_t)blockIdx.x * 8 + wave;

  // Stage this row's 64 E values in LDS for the per-lane gather.
  eRow[wave][lane]      = E[row * TT + lane];
  eRow[wave][lane + 32] = E[row * TT + lane + 32];
  __syncthreads();

  const float* src = qk + row * (size_t)NN;
  float* dst       = out + row * (size_t)NN;

  float carry = 0.0f;
#pragma unroll 1
  for (int k = 0; k < NN / 32; ++k) {
    // Reversed chunk: lane 0 holds the highest j -> inclusive lane scan
    // yields the suffix sum. Each chunk is one contiguous 128B segment.
    const int j = (NN - 1) - k * 32 - lane;
    const float x = __builtin_nontemporal_load(src + j);
    const float g = 1.0f / (1.0f + __expf(-x));

    float v = g;
#pragma unroll
    for (int off = 1; off < 32; off <<= 1) {
      const float t = __shfl_up(v, off, 32);
      if (lane >= off) v += t;
    }
    const float P = carry + v;
    carry = __shfl(P, 31, 32);  // lane 31 holds carry + chunk total

    const float Pc = fminf(P, (float)(TT - 1));
    const float pf = floorf(Pc);
    const float w = Pc - pf;
    const int i0 = (int)pf;
    int i1 = i0 + 1;
    if (i1 > TT - 1) i1 = TT - 1;  // w==0 there, so value matches reference

    const float ef = eRow[wave][i0];
    const float ec = eRow[wave][i1];
    __builtin_nontemporal_store(fmaf(w, ec - ef, ef), dst + j);
  }
}

// ---------------------------------------------------------------------------
extern "C" void kernel_launch(void* const* d_in, const int* in_sizes, int n_in,
                              void* d_out, int out_size, void* d_ws,
                              size_t ws_size, hipStream_t stream) {
  (void)in_sizes; (void)n_in; (void)out_size; (void)ws_size;
  const float* q  = (const float*)d_in[0];   // (BH, NN, CC)
  const float* qk = (const float*)d_in[1];   // (BH, NN, NN)
  const float* pe = (const float*)d_in[2];   // (1, CC, TT)
  float* out = (float*)d_out;                // (BH, NN, NN)
  float* E   = (float*)d_ws;                 // (BH*NN, TT) = 8 MB scratch

  // Kernel 1: 2048 row-tiles, 8 waves/block -> 256 blocks.
  cope_e_wmma<<<dim3((BH * NN / 16) / 8), dim3(256), 0, stream>>>(q, pe, E);
  // Kernel 2: 32768 rows, 8 waves/block -> 4096 blocks.
  cope_scan<<<dim3((BH * NN) / 8), dim3(256), 0, stream>>>(qk, E, out);
}